// model003_37666863186444
// MI455X (gfx1250) — compile-verified
//
#include <hip/hip_runtime.h>
#include <stdint.h>

typedef __attribute__((ext_vector_type(2))) float v2f;
typedef __attribute__((ext_vector_type(8))) float v8f;

#define NYD    64
#define NUD    64
#define NHD    256
#define NPD    128
#define NT     1024
#define NBATCH 128
#define CHUNK  16
#define NCHUNK 64     // NT / CHUNK
#define BTILE  16
#define NBT    8      // NBATCH / BTILE

// ---- workspace layout (float offsets) ----
enum : size_t {
  WS_LR    = 0,
  WS_LI    = WS_LR  + NPD,
  WS_LRL   = WS_LI  + NPD,
  WS_LIL   = WS_LRL + NPD,
  WS_POW   = WS_LIL + NPD,                        // [CHUNK][NPD][2]  Λ^q, q=1..16
  WS_BT    = WS_POW + (size_t)CHUNK * NPD * 2,    // [64][256]  B^T scaled by nf
  WS_WT    = WS_BT  + (size_t)64 * 256,           // [256][64]  W_x2y^T
  WS_X0    = WS_WT  + (size_t)256 * 64,           // [128][256]
  WS_CARRY = WS_X0  + (size_t)128 * 256,          // [NCHUNK][128][256]
  WS_OFF   = WS_CARRY + (size_t)NCHUNK * 128 * 256,
};

__device__ inline v8f wmma_f32x4(v2f a, v2f b, v8f c) {
  return __builtin_amdgcn_wmma_f32_16x16x4_f32(false, a, false, b, (short)0, c,
                                               false, false);
}

// ---------- prep: lambda params + Λ^q table ----------
__global__ __launch_bounds__(128) void prep_lambda_kernel(
    const float* __restrict__ lcr, const float* __restrict__ lci,
    float* __restrict__ w) {
  int p = threadIdx.x;  // 0..127
  float a  = fabsf(lcr[p]);
  float th = 1.57079632679489662f * lci[p];
  float r  = __expf(-a);
  w[WS_LR + p] = r * __cosf(th);
  w[WS_LI + p] = r * __sinf(th);
  float rL = __expf(-(float)CHUNK * a);
  w[WS_LRL + p] = rL * __cosf((float)CHUNK * th);
  w[WS_LIL + p] = rL * __sinf((float)CHUNK * th);
#pragma unroll
  for (int q = 1; q <= CHUNK; ++q) {
    float rq = __expf(-(float)q * a);
    w[WS_POW + ((size_t)(q - 1) * NPD + p) * 2 + 0] = rq * __cosf((float)q * th);
    w[WS_POW + ((size_t)(q - 1) * NPD + p) * 2 + 1] = rq * __sinf((float)q * th);
  }
}

// ---------- prep: Bt[k][n] = B[n][k] * nf[n] ----------
__global__ __launch_bounds__(256) void prep_bt_kernel(
    const float* __restrict__ B, const float* __restrict__ lcr,
    float* __restrict__ w) {
  int idx = blockIdx.x * 256 + threadIdx.x;  // 0..16383
  int k = idx >> 8, n = idx & 255, p = n & 127;
  float nf = sqrtf(1.f - __expf(-2.f * fabsf(lcr[p])));
  w[WS_BT + idx] = B[n * NUD + k] * nf;
}

// ---------- prep: Wt[k][n] = W_x2y[n][k] ----------
__global__ __launch_bounds__(256) void prep_wt_kernel(
    const float* __restrict__ Wxy, float* __restrict__ w) {
  int idx = blockIdx.x * 256 + threadIdx.x;  // 0..16383
  int k = idx >> 6, n = idx & 63;
  w[WS_WT + idx] = Wxy[n * NHD + k];
}

// ---------- prep: x0 = y0 @ W_y2x^T + b_y2x ----------
__global__ __launch_bounds__(256) void prep_x0_kernel(
    const float* __restrict__ y0, const float* __restrict__ Wyx,
    const float* __restrict__ byx, float* __restrict__ w) {
  int idx = blockIdx.x * 256 + threadIdx.x;  // 0..32767
  int b = idx >> 8, h = idx & 255;
  float acc = byx[h];
#pragma unroll 8
  for (int y = 0; y < NYD; ++y) acc += y0[b * NYD + y] * Wyx[h * NYD + y];
  w[WS_X0 + idx] = acc;
}

// ---------- fused Bu-GEMM + local chunk scan + output GEMM ----------
// grid (NCHUNK, NBT), block 128 (4 waves).
// Hidden-state tiles per wave: cols {32w, 32w+16, 128+32w, 128+32w+16}
// so rotation pairs (h, h+128) live in the same lane/slot.
// Y columns per wave: [16w, 16w+16).
// U tiles are prefetched with async global->LDS into per-wave double buffers.
__global__ __launch_bounds__(128) void bu_scan_kernel(
    const float* __restrict__ U, const float* __restrict__ bxy,
    float* __restrict__ w, float* __restrict__ out) {
  __shared__ float sU[4][2][16][68];  // per-wave double-buffered U tiles
  __shared__ float sX[16][260];       // shared state tile for output GEMM

  const int tid = threadIdx.x;
  const int wid = tid >> 5, lane = tid & 31, hl = lane >> 4, lm = lane & 15;
  const int c = blockIdx.x, bbase = blockIdx.y * BTILE;

  int n0[4];
  n0[0] = wid * 32;     n0[1] = wid * 32 + 16;
  n0[2] = n0[0] + 128;  n0[3] = n0[1] + 128;
  const int n0y = wid * 16;

  // hoist loop-invariant B^T*nf fragments into registers (16 kk x 4 tiles)
  v2f bf[16][4];
#pragma unroll
  for (int kk = 0; kk < 16; ++kk) {
    const int k0 = kk * 4 + hl * 2;
#pragma unroll
    for (int j = 0; j < 4; ++j) {
      bf[kk][j].x = w[WS_BT + (size_t)k0 * NHD + n0[j] + lm];
      bf[kk][j].y = w[WS_BT + (size_t)(k0 + 1) * NHD + n0[j] + lm];
    }
  }
  // hoist loop-invariant W^T fragments (64 kk x 1 Y-tile per wave)
  v2f wf[64];
#pragma unroll
  for (int kk = 0; kk < 64; ++kk) {
    const int k0 = kk * 4 + hl * 2;
    wf[kk].x = w[WS_WT + (size_t)k0 * NYD + n0y + lm];
    wf[kk].y = w[WS_WT + (size_t)(k0 + 1) * NYD + n0y + lm];
  }

  const float lr0 = w[WS_LR + n0[0] + lm], li0 = w[WS_LI + n0[0] + lm];
  const float lr1 = w[WS_LR + n0[1] + lm], li1 = w[WS_LI + n0[1] + lm];
  const float bias = bxy[n0y + lm];

  v8f st[4];
#pragma unroll
  for (int j = 0; j < 4; ++j) st[j] = (v8f)0.f;

  // per-wave async prefetch of one 16x64 U tile (8 x b128 per lane)
  auto issueU = [&](int t, int buf) {
#pragma unroll
    for (int e = 0; e < 8; ++e) {
      const int i = lane + e * 32;  // 0..255 float4 slots
      const int row = i >> 4, c4 = (i & 15) * 4;
      uint32_t lds = (uint32_t)(uintptr_t)&sU[wid][buf][row][c4];
      uint32_t go  = (uint32_t)(((uint32_t)(t * NBATCH + bbase + row) * NUD + c4) * 4u);
      asm volatile("global_load_async_to_lds_b128 %0, %1, %2"
                   :: "v"(lds), "v"(go), "s"(U) : "memory");
    }
  };

  issueU(c * CHUNK, 0);

  for (int tt = 0; tt < CHUNK; ++tt) {
    const int t = c * CHUNK + tt;
    if (tt + 1 < CHUNK) {
      issueU(t + 1, (tt + 1) & 1);
      asm volatile("s_wait_asynccnt 0x8" ::: "memory");  // current tile landed
    } else {
      asm volatile("s_wait_asynccnt 0x0" ::: "memory");
    }
    const float (*cu)[68] = sU[wid][tt & 1];

    // bu = U_tile @ (B^T * nf)
    v8f acc[4];
#pragma unroll
    for (int j = 0; j < 4; ++j) acc[j] = (v8f)0.f;
#pragma unroll
    for (int kk = 0; kk < 16; ++kk) {
      const int k0 = kk * 4 + hl * 2;
      v2f a = *(const v2f*)&cu[lm][k0];
#pragma unroll
      for (int j = 0; j < 4; ++j) acc[j] = wmma_f32x4(a, bf[kk][j], acc[j]);
    }

    // x' = Λ x + bu  (per-lane complex rotate; pairs co-located)
#pragma unroll
    for (int j = 0; j < 2; ++j) {
      const float lr = (j == 0) ? lr0 : lr1;
      const float li = (j == 0) ? li0 : li1;
#pragma unroll
      for (int r = 0; r < 8; ++r) {
        float x1 = st[j][r], x2 = st[j + 2][r];
        st[j][r]     = lr * x1 - li * x2 + acc[j][r];
        st[j + 2][r] = li * x1 + lr * x2 + acc[j + 2][r];
      }
    }

    // share state tile, then Y_partial[t] = x~ @ W^T + bias
#pragma unroll
    for (int j = 0; j < 4; ++j) {
      const int col = n0[j] + lm;
#pragma unroll
      for (int r = 0; r < 8; ++r) sX[hl * 8 + r][col] = st[j][r];
    }
    __syncthreads();

    v8f ya = (v8f)0.f;
#pragma unroll
    for (int kk = 0; kk < 64; ++kk) {
      const int k0 = kk * 4 + hl * 2;
      v2f a = *(const v2f*)&sX[lm][k0];
      ya = wmma_f32x4(a, wf[kk], ya);
    }
#pragma unroll
    for (int r = 0; r < 8; ++r) {
      size_t gi = ((size_t)t * NBATCH + bbase + hl * 8 + r) * NYD + n0y + lm;
      out[gi] = ya[r] + bias;
    }
    __syncthreads();  // sX reused next timestep
  }

  // chunk carry = final local state
#pragma unroll
  for (int j = 0; j < 4; ++j) {
    const int col = n0[j] + lm;
#pragma unroll
    for (int r = 0; r < 8; ++r) {
      size_t gi = ((size_t)c * NBATCH + bbase + hl * 8 + r) * NHD + col;
      w[WS_CARRY + gi] = st[j][r];
    }
  }
}

// ---------- sequential carry combine (the only serial part) ----------
__global__ __launch_bounds__(256) void combine_kernel(float* __restrict__ w) {
  int idx = blockIdx.x * 256 + threadIdx.x;  // 0..16383
  int b = idx >> 7, p = idx & 127;
  float o1 = w[WS_X0 + (size_t)b * NHD + p];
  float o2 = w[WS_X0 + (size_t)b * NHD + p + 128];
  const float lrL = w[WS_LRL + p], liL = w[WS_LIL + p];
  for (int c = 0; c < NCHUNK; ++c) {
    size_t base = ((size_t)c * NBATCH + b) * NHD;
    w[WS_OFF + base + p]       = o1;
    w[WS_OFF + base + p + 128] = o2;
    float c1 = w[WS_CARRY + base + p];
    float c2 = w[WS_CARRY + base + p + 128];
    float t1 = lrL * o1 - liL * o2 + c1;
    float t2 = liL * o1 + lrL * o2 + c2;
    o1 = t1; o2 = t2;
  }
}

// ---------- correction: Y[t] += (Λ^{tt+1} offset_c) @ W^T ----------
// grid (NCHUNK, NBT), block 128 (4 waves); no X~ tensor needed at all.
__global__ __launch_bounds__(128) void corr_kernel(
    const float* __restrict__ w, float* __restrict__ out) {
  __shared__ float sA[16][260];
  __shared__ float sPow[CHUNK][NPD][2];  // Λ^q table
  __shared__ float sOff[16][256];        // this block's offset tile

  const int tid = threadIdx.x;
  const int wid = tid >> 5, lane = tid & 31, hl = lane >> 4, lm = lane & 15;
  const int c = blockIdx.x, bbase = blockIdx.y * BTILE;
  const int n0y = wid * 16;

  v2f wf[64];
#pragma unroll
  for (int kk = 0; kk < 64; ++kk) {
    const int k0 = kk * 4 + hl * 2;
    wf[kk].x = w[WS_WT + (size_t)k0 * NYD + n0y + lm];
    wf[kk].y = w[WS_WT + (size_t)(k0 + 1) * NYD + n0y + lm];
  }

  const float4* gp = (const float4*)(w + WS_POW);
  float* sp = &sPow[0][0][0];
#pragma unroll 4
  for (int i = tid; i < 1024; i += 128) *(float4*)&sp[i * 4] = gp[i];
  const float4* go = (const float4*)(w + WS_OFF + ((size_t)c * NBATCH + bbase) * NHD);
  float* so = &sOff[0][0];
#pragma unroll 4
  for (int i = tid; i < 1024; i += 128) *(float4*)&so[i * 4] = go[i];
  __syncthreads();

  for (int tt = 0; tt < CHUNK; ++tt) {
    const int t = c * CHUNK + tt;
    // A = Λ^{tt+1} * offset  (pure closed-form, no scan tensor)
#pragma unroll
    for (int e = 0; e < 32; ++e) {
      int idx = tid + e * 128;  // 0..4095
      int row = idx >> 8, h = idx & 255, p = h & 127;
      float cq = sPow[tt][p][0], sq = sPow[tt][p][1];
      float o1 = sOff[row][p], o2 = sOff[row][p + 128];
      sA[row][h] = (h < 128) ? (cq * o1 - sq * o2) : (sq * o1 + cq * o2);
    }
    __syncthreads();

    v8f acc = (v8f)0.f;
#pragma unroll
    for (int kk = 0; kk < 64; ++kk) {
      const int k0 = kk * 4 + hl * 2;
      v2f a = *(const v2f*)&sA[lm][k0];
      acc = wmma_f32x4(a, wf[kk], acc);
    }
#pragma unroll
    for (int r = 0; r < 8; ++r) {
      size_t gi = ((size_t)t * NBATCH + bbase + hl * 8 + r) * NYD + n0y + lm;
      out[gi] += acc[r];
    }
    __syncthreads();
  }
}

extern "C" void kernel_launch(void* const* d_in, const int* in_sizes, int n_in,
                              void* d_out, int out_size, void* d_ws, size_t ws_size,
                              hipStream_t stream) {
  const float* y0  = (const float*)d_in[0];
  const float* U   = (const float*)d_in[1];
  const float* lcr = (const float*)d_in[2];
  const float* lci = (const float*)d_in[3];
  const float* B   = (const float*)d_in[4];
  const float* Wyx = (const float*)d_in[5];
  const float* byx = (const float*)d_in[6];
  const float* Wxy = (const float*)d_in[7];
  const float* bxy = (const float*)d_in[8];
  float* out = (float*)d_out;
  float* w   = (float*)d_ws;

  prep_lambda_kernel<<<1, 128, 0, stream>>>(lcr, lci, w);
  prep_bt_kernel<<<64, 256, 0, stream>>>(B, lcr, w);
  prep_wt_kernel<<<64, 256, 0, stream>>>(Wxy, w);
  prep_x0_kernel<<<128, 256, 0, stream>>>(y0, Wyx, byx, w);
  bu_scan_kernel<<<dim3(NCHUNK, NBT), 128, 0, stream>>>(U, bxy, w, out);
  combine_kernel<<<64, 256, 0, stream>>>(w);
  corr_kernel<<<dim3(NCHUNK, NBT), 128, 0, stream>>>(w, out);
}